// LocalPredictiveAttention_10969346474151
// MI455X (gfx1250) — compile-verified
//
#include <hip/hip_runtime.h>

// ---------------------------------------------------------------------------
// Local predictive attention, MI455X (gfx1250, wave32).
//  K1: Wph = tanh(hidden @ Wp_w^T + Wp_b) via V_WMMA_F32_16X16X4_F32
//  K2: per-batch fused p / window scores / masked softmax / gaussian / context
// Bandwidth-bound problem (~18 MB total); window re-read in phase C hits the
// 192 MB L2, so no giant LDS staging is needed.
// ---------------------------------------------------------------------------

typedef __attribute__((ext_vector_type(2))) float v2f;
typedef __attribute__((ext_vector_type(8))) float v8f;

#define WIN_D 64
#define WIN_W 129   // 2*WIN_D + 1
#define HMAX  512

// ---------------------------------------------------------------------------
// Kernel 1: one wave (32 threads) per 16x16 output tile of Wph (B x H).
// D = A(16x4) x B(4x16) + C accumulated over K = H in steps of 4, f32 WMMA.
//   A(m,k) = hidden[m*H + k]
//   B(k,n) = Wp_w[n*H + k]   (hidden @ Wp_w^T)
// Fragment layout (ISA 7.12.2, 32-bit A 16x4): lane<16 -> M=lane, holds
// K = {k0, k0+1}; lane>=16 -> M=lane-16, holds K = {k0+2, k0+3}. B mirrored
// with N striped on lanes. C/D: VGPR r, lanes 0-15 -> (M=r, N=lane),
// lanes 16-31 -> (M=8+r, N=lane-16).
// ---------------------------------------------------------------------------
__global__ void lpa_proj_tanh_wmma(const float* __restrict__ hidden,
                                   const float* __restrict__ Wp_w,
                                   const float* __restrict__ Wp_b,
                                   float* __restrict__ Wph,
                                   int B, int H) {
    const int lane  = threadIdx.x;           // 0..31 (wave32)
    const int tileN = blockIdx.x;            // column tile of Wph (H/16)
    const int tileM = blockIdx.y;            // row tile of Wph    (B/16)
    const int half  = lane >> 4;             // 0: lanes 0-15, 1: lanes 16-31
    const int l16   = lane & 15;

    const float* aRow = hidden + (size_t)(tileM * 16 + l16) * H; // A row (M)
    const float* bRow = Wp_w   + (size_t)(tileN * 16 + l16) * H; // B row (N)

    v8f acc = {};
    for (int k0 = 0; k0 < H; k0 += 4) {
        const int kk = k0 + 2 * half;        // lane-half K offset
        v2f a = *(const v2f*)(aRow + kk);    // A(m, kk), A(m, kk+1)
        v2f bf = *(const v2f*)(bRow + kk);   // B(kk, n), B(kk+1, n)
        // (neg_a, A, neg_b, B, c_mod, C, reuse_a, reuse_b)
        acc = __builtin_amdgcn_wmma_f32_16x16x4_f32(
            false, a, false, bf, (short)0, acc, false, false);
    }

    const int nG = tileN * 16 + l16;
    const float bias = Wp_b[nG];
#pragma unroll
    for (int r = 0; r < 8; ++r) {
        const int mG = tileM * 16 + half * 8 + r;
        Wph[(size_t)mG * H + nG] = tanhf(acc[r] + bias);
    }
}

// ---------------------------------------------------------------------------
// Kernel 2: one block (256 threads = 8 waves) per batch element b.
//  0) p[b] = S * sigmoid(dot(Wph[b], vp_w) + vp_b); center = rint(p)
//  1) scores[wi] = dot(hidden[b], enc[center-D+wi, b, :]) (masked -> -1e9)
//     (one wave per window position, wave32 xor-shuffle reduction)
//  2) masked softmax over 129, times gaussian around unrounded p
//  3) context[b,h] = sum_wi attn[wi] * enc[clamp(idx), b, h]  (L2 re-read)
// ---------------------------------------------------------------------------
__global__ void lpa_attention(const float* __restrict__ hidden,
                              const float* __restrict__ enc,
                              const float* __restrict__ vp_w,
                              const float* __restrict__ vp_b,
                              const float* __restrict__ Wph,
                              float* __restrict__ attn_out,
                              float* __restrict__ ctx_out,
                              int B, int S, int H) {
    const int b = blockIdx.x;
    const int t = threadIdx.x;               // 0..255

    __shared__ float s_hidden[HMAX];
    __shared__ float s_tmp[256];
    __shared__ float s_attn[WIN_W];
    __shared__ float s_p;
    __shared__ int   s_center;

    // ---- load hidden row to LDS + partial dot for p -----------------------
    float pd = 0.f;
    for (int h = t; h < H; h += 256) {
        float hv = hidden[(size_t)b * H + h];
        s_hidden[h] = hv;
        pd += Wph[(size_t)b * H + h] * vp_w[h];
    }
    s_tmp[t] = pd;
    __syncthreads();
    for (int off = 128; off > 0; off >>= 1) {
        if (t < off) s_tmp[t] += s_tmp[t + off];
        __syncthreads();
    }
    if (t == 0) {
        float x = s_tmp[0] + vp_b[0];
        float p = (float)S * (1.f / (1.f + expf(-x)));
        s_p = p;
        s_center = (int)rintf(p);            // round-half-even, matches jnp.round
    }
    __syncthreads();
    const float p      = s_p;
    const int   center = s_center;

    // ---- phase A: window scores (one wave per window index) ---------------
    const int wave = t >> 5;
    const int lane = t & 31;
    for (int wi = wave; wi < WIN_W; wi += 8) {
        const int sIdx = center - WIN_D + wi;
        float score = -1e9f;
        if (sIdx >= 0 && sIdx < S) {
            const float* row = enc + ((size_t)sIdx * B + b) * H;
            // prefetch the row this wave will touch next iteration
            const int sNext = sIdx + 8;
            if (sNext >= 0 && sNext < S)
                __builtin_prefetch(enc + ((size_t)sNext * B + b) * H + lane * 4, 0, 0);
            float acc = 0.f;
            for (int h = lane; h < H; h += 32)
                acc += s_hidden[h] * row[h];
#pragma unroll
            for (int off = 16; off > 0; off >>= 1)
                acc += __shfl_xor(acc, off, 32);
            score = acc;
        }
        if (lane == 0) s_attn[wi] = score;
    }
    __syncthreads();

    // ---- phase B: masked softmax * gaussian -------------------------------
    float sc = (t < WIN_W) ? s_attn[t] : -3.4e38f;
    s_tmp[t] = sc;
    __syncthreads();
    for (int off = 128; off > 0; off >>= 1) {
        if (t < off) s_tmp[t] = fmaxf(s_tmp[t], s_tmp[t + off]);
        __syncthreads();
    }
    const float mx = s_tmp[0];
    __syncthreads();
    const float e = (t < WIN_W) ? expf(sc - mx) : 0.f;
    s_tmp[t] = e;
    __syncthreads();
    for (int off = 128; off > 0; off >>= 1) {
        if (t < off) s_tmp[t] += s_tmp[t + off];
        __syncthreads();
    }
    const float denom = s_tmp[0];
    __syncthreads();
    if (t < WIN_W) {
        const float stddev = (float)WIN_D * 0.5f;                // 32
        const float d  = (float)(center - WIN_D + t) - p;        // unrounded p
        const float g  = expf(-(d * d) / (2.f * stddev * stddev));
        const float a  = (e / denom) * g;
        s_attn[t] = a;
        attn_out[(size_t)b * WIN_W + t] = a;
    }
    __syncthreads();

    // ---- phase C: context (window rows now resident in 192MB L2) ----------
    for (int h = t; h < H; h += 256) {
        float acc = 0.f;
        for (int wi = 0; wi < WIN_W; ++wi) {
            int sIdx = center - WIN_D + wi;
            int scl  = sIdx < 0 ? 0 : (sIdx >= S ? S - 1 : sIdx);
            acc += s_attn[wi] * enc[((size_t)scl * B + b) * H + h];
        }
        ctx_out[(size_t)b * H + h] = acc;
    }
}

// ---------------------------------------------------------------------------
extern "C" void kernel_launch(void* const* d_in, const int* in_sizes, int n_in,
                              void* d_out, int out_size, void* d_ws, size_t ws_size,
                              hipStream_t stream) {
    // inputs: 0:t(int) 1:hidden(B,H) 2:enc(S,B,H) 3:Wp_w(H,H) 4:Wp_b(H)
    //         5:vp_w(H) 6:vp_b(1)
    const float* hidden = (const float*)d_in[1];
    const float* enc    = (const float*)d_in[2];
    const float* Wp_w   = (const float*)d_in[3];
    const float* Wp_b   = (const float*)d_in[4];
    const float* vp_w   = (const float*)d_in[5];
    const float* vp_b   = (const float*)d_in[6];

    const int H = in_sizes[4];                 // 512
    const int B = in_sizes[1] / H;             // 64
    const int S = in_sizes[2] / in_sizes[1];   // 4096

    float* Wph      = (float*)d_ws;            // B*H floats of scratch
    float* attn_out = (float*)d_out;                     // (B, 129)
    float* ctx_out  = attn_out + (size_t)B * WIN_W;      // (B, H)

    dim3 g1(H / 16, B / 16);                   // (32, 4) tiles, 1 wave each
    lpa_proj_tanh_wmma<<<g1, 32, 0, stream>>>(hidden, Wp_w, Wp_b, Wph, B, H);
    lpa_attention<<<B, 256, 0, stream>>>(hidden, enc, vp_w, vp_b, Wph,
                                         attn_out, ctx_out, B, S, H);
}